// NatUNet_2834678415879
// MI455X (gfx1250) — compile-verified
//
#include <hip/hip_runtime.h>
#include <cmath>

#define KNB 8
#define NCLS 21

typedef __attribute__((ext_vector_type(2))) float v2f;
typedef __attribute__((ext_vector_type(8))) float v8f;

// ---------------------------------------------------------------------------
// Weight repack for WMMA-native B fetches:
//   Wp[((k/4)*2 + ((k%4)>>1))*N + n] (float2, element k%2) = W[k][n]
// After this, a lane's two K-rows for its column are one contiguous float2.
// ---------------------------------------------------------------------------
__global__ void repack_w_kernel(const float* __restrict__ W,
                                float* __restrict__ Wp, int N, int K) {
  int total = K * N;
  int idx = blockIdx.x * blockDim.x + threadIdx.x;
  if (idx >= total) return;
  int n = idx % N;
  int k = idx / N;
  int g = k >> 2;
  int h = (k >> 1) & 1;
  int e = k & 1;
  Wp[(((size_t)(g * 2 + h)) * N + n) * 2 + e] = W[(size_t)k * N + n];
}

// ---------------------------------------------------------------------------
// Generic GEMM via V_WMMA_F32_16X16X4_F32 (fp32 in, fp32 accumulate).
// C[M,N] = A[M,K] @ W[K,N] (+ bias[N]), W pre-packed by repack_w_kernel.
// One wave computes a 16x32 C tile (two 16x16 WMMA tiles sharing the A fetch).
// K unrolled by 2 (K%4==0 required; K%8==0 fast path, 4-wide tail).
// Out-of-range rows/cols are CLAMPED on load (they only feed accumulator
// entries that are never stored), so EXEC stays all-ones for the WMMAs.
// A 16x4 lane layout: lanes 0-15 -> M=lane, K+{0,1}; lanes 16-31 -> K+{2,3}.
// B 4x16:             lanes 0-15 -> N=lane, rows K+{0,1}; lanes 16-31 -> K+{2,3}.
// C/D: VGPR j -> (M = j + 8*(lane>=16), N = lane%16).
// All A and B fetches are aligned global_load_b64.
// ---------------------------------------------------------------------------
__global__ __launch_bounds__(256) void gemm_wmma_f32_kernel(
    const float* __restrict__ A, const float* __restrict__ Wp,
    const float* __restrict__ bias, float* __restrict__ C,
    int M, int N, int K) {
  int gwave = (int)((blockIdx.x * (size_t)blockDim.x + threadIdx.x) >> 5);
  int lane  = threadIdx.x & 31;
  int tilesN = (N + 31) >> 5;   // 32-wide N tiles
  int tilesM = (M + 15) >> 4;
  if (gwave >= tilesM * tilesN) return;          // wave-uniform exit
  int tm = (gwave / tilesN) << 4;
  int tn = (gwave % tilesN) << 5;
  int half = lane >> 4;
  int l16  = lane & 15;

  int arow = tm + l16;
  int arowc = arow < M ? arow : (M - 1);         // clamped load row
  int nc0 = tn + l16;
  int nc1 = tn + 16 + l16;
  int nc0c = nc0 < N ? nc0 : (N - 1);
  int nc1c = nc1 < N ? nc1 : (N - 1);

  const float* ap  = A + (size_t)arowc * K + (half << 1);
  const float* bp0 = Wp + ((size_t)half * N + nc0c) * 2;
  const float* bp1 = Wp + ((size_t)half * N + nc1c) * 2;
  const size_t bstep = (size_t)4 * N;            // floats per K-group of 4

  v8f acc0 = {};
  v8f acc1 = {};

  int k0 = 0;
  for (; k0 + 8 <= K; k0 += 8) {
    v2f a0  = *(const v2f*)(ap);
    v2f a1  = *(const v2f*)(ap + 4);
    v2f b00 = *(const v2f*)(bp0);
    v2f b01 = *(const v2f*)(bp1);
    v2f b10 = *(const v2f*)(bp0 + bstep);
    v2f b11 = *(const v2f*)(bp1 + bstep);
    acc0 = __builtin_amdgcn_wmma_f32_16x16x4_f32(false, a0, false, b00,
                                                 (short)0, acc0, false, false);
    acc1 = __builtin_amdgcn_wmma_f32_16x16x4_f32(false, a0, false, b01,
                                                 (short)0, acc1, false, false);
    acc0 = __builtin_amdgcn_wmma_f32_16x16x4_f32(false, a1, false, b10,
                                                 (short)0, acc0, false, false);
    acc1 = __builtin_amdgcn_wmma_f32_16x16x4_f32(false, a1, false, b11,
                                                 (short)0, acc1, false, false);
    ap  += 8;
    bp0 += 2 * bstep;
    bp1 += 2 * bstep;
  }
  if (k0 < K) {                                   // K%8==4 tail
    v2f a0  = *(const v2f*)(ap);
    v2f b00 = *(const v2f*)(bp0);
    v2f b01 = *(const v2f*)(bp1);
    acc0 = __builtin_amdgcn_wmma_f32_16x16x4_f32(false, a0, false, b00,
                                                 (short)0, acc0, false, false);
    acc1 = __builtin_amdgcn_wmma_f32_16x16x4_f32(false, a0, false, b01,
                                                 (short)0, acc1, false, false);
  }

  if (nc0 < N) {
    float bb = bias ? bias[nc0] : 0.0f;
#pragma unroll
    for (int j = 0; j < 8; ++j) {
      int r = tm + j + (half << 3);
      if (r < M) C[(size_t)r * N + nc0] = acc0[j] + bb;
    }
  }
  if (nc1 < N) {
    float bb = bias ? bias[nc1] : 0.0f;
#pragma unroll
    for (int j = 0; j < 8; ++j) {
      int r = tm + j + (half << 3);
      if (r < M) C[(size_t)r * N + nc1] = acc1[j] + bb;
    }
  }
}

// ---------------------------------------------------------------------------
// Patch embed: x(B,3,224,224) -> z(B,56,56,96), 4x4 patches.
// ---------------------------------------------------------------------------
__global__ void patch_embed_kernel(const float* __restrict__ X,
                                   const float* __restrict__ Wt,
                                   const float* __restrict__ bias,
                                   float* __restrict__ Z, int Bz) {
  const int H = 56, W = 56, C0 = 96, P = 4, CH = 3, HIN = 224;
  int total = Bz * H * W * C0;
  int idx = blockIdx.x * blockDim.x + threadIdx.x;
  if (idx >= total) return;
  int o = idx % C0;
  int t = idx / C0;
  int w = t % W; t /= W;
  int h = t % H;
  int b = t / H;
  float acc = bias[o];
  for (int c = 0; c < CH; ++c)
    for (int p = 0; p < P; ++p)
      for (int q = 0; q < P; ++q)
        acc += X[(((size_t)b * CH + c) * HIN + (h * P + p)) * HIN + (w * P + q)] *
               Wt[(((size_t)o * CH + c) * P + p) * P + q];
  Z[idx] = acc;
}

// ---------------------------------------------------------------------------
// Neighborhood attention pieces (window KNB=8 with wrap over N=H*W per batch).
// ---------------------------------------------------------------------------
__global__ void na_scores_kernel(const float* __restrict__ Q,
                                 const float* __restrict__ Kt,
                                 const float* __restrict__ rb,
                                 float* __restrict__ S,
                                 int Bz, int N, int C, float scale) {
  int total = Bz * N * KNB;
  int idx = blockIdx.x * blockDim.x + threadIdx.x;
  if (idx >= total) return;
  int k = idx % KNB;
  int t = idx / KNB;
  int n = t % N;
  int b = t / N;
  int nb = (n + k) % N;
  const float* q  = Q  + ((size_t)b * N + n)  * C;
  const float* kk = Kt + ((size_t)b * N + nb) * C;
  float s = 0.0f;
  for (int c = 0; c < C; ++c) s += q[c] * kk[c];
  S[idx] = (s + rb[k]) * scale;   // rb row 0, then /sqrt(C)
}

__global__ void na_softmax_kernel(float* __restrict__ S, int tokens) {
  int n = blockIdx.x * blockDim.x + threadIdx.x;
  if (n >= tokens) return;
  float* s = S + (size_t)n * KNB;
  float m = s[0];
#pragma unroll
  for (int k = 1; k < KNB; ++k) m = fmaxf(m, s[k]);
  float sum = 0.0f;
#pragma unroll
  for (int k = 0; k < KNB; ++k) { float e = __expf(s[k] - m); s[k] = e; sum += e; }
  float r = 1.0f / sum;
#pragma unroll
  for (int k = 0; k < KNB; ++k) s[k] *= r;
}

__global__ void na_out_kernel(const float* __restrict__ Pm,
                              const float* __restrict__ V,
                              float* __restrict__ O,
                              int Bz, int N, int C) {
  int total = Bz * N * C;
  int idx = blockIdx.x * blockDim.x + threadIdx.x;
  if (idx >= total) return;
  int c = idx % C;
  int t = idx / C;
  int n = t % N;
  int b = t / N;
  const float* p = Pm + ((size_t)b * N + n) * KNB;
  float acc = 0.0f;
#pragma unroll
  for (int k = 0; k < KNB; ++k) {
    int nb = (n + k) % N;
    acc += p[k] * V[((size_t)b * N + nb) * C + c];
  }
  O[idx] = acc;
}

// ---------------------------------------------------------------------------
// Patch merge: gather 2x2 (order [00,01,10,11]) + LayerNorm(4C).
// ---------------------------------------------------------------------------
__global__ void merge_ln_kernel(const float* __restrict__ Z,
                                const float* __restrict__ g,
                                const float* __restrict__ bt,
                                float* __restrict__ Y,
                                int Bz, int H, int W, int C) {
  int H2 = H >> 1, W2 = W >> 1;
  int total = Bz * H2 * W2;
  int idx = blockIdx.x * blockDim.x + threadIdx.x;
  if (idx >= total) return;
  int w2 = idx % W2;
  int t = idx / W2;
  int h2 = t % H2;
  int b = t / H2;
  const float* p[4];
  p[0] = Z + (((size_t)b * H + 2 * h2) * W + 2 * w2) * C;
  p[1] = p[0] + C;
  p[2] = Z + (((size_t)b * H + 2 * h2 + 1) * W + 2 * w2) * C;
  p[3] = p[2] + C;
  int C4 = 4 * C;
  float mean = 0.0f;
  for (int s = 0; s < 4; ++s)
    for (int c = 0; c < C; ++c) mean += p[s][c];
  mean /= (float)C4;
  float var = 0.0f;
  for (int s = 0; s < 4; ++s)
    for (int c = 0; c < C; ++c) { float d = p[s][c] - mean; var += d * d; }
  var /= (float)C4;
  float r = rsqrtf(var + 1e-5f);
  float* y = Y + (size_t)idx * C4;
  for (int s = 0; s < 4; ++s)
    for (int c = 0; c < C; ++c) {
      int j = s * C + c;
      y[j] = (p[s][c] - mean) * r * g[j] + bt[j];
    }
}

// ---------------------------------------------------------------------------
// Decoder expand: ZW(B,H,W,2C) -> pixel-unshuffle to (B,2H,2W,C/2), LN(C/2),
// then concat skip(B,2H,2W,C/2) -> Y(B,2H,2W,C).
// ---------------------------------------------------------------------------
__global__ void expand_ln_concat_kernel(const float* __restrict__ ZW,
                                        const float* __restrict__ g,
                                        const float* __restrict__ bt,
                                        const float* __restrict__ SK,
                                        float* __restrict__ Y,
                                        int Bz, int H, int W, int C) {
  int Ch = C >> 1;
  int H2 = H << 1, W2 = W << 1;
  int total = Bz * H2 * W2;
  int idx = blockIdx.x * blockDim.x + threadIdx.x;
  if (idx >= total) return;
  int x = idx % W2;
  int t = idx / W2;
  int y = t % H2;
  int b = t / H2;
  int h = y >> 1, a = y & 1, w = x >> 1, c2 = x & 1;
  const float* src = ZW + (((size_t)b * H + h) * W + w) * (2 * C) +
                     (size_t)a * C + (size_t)c2 * Ch;
  float mean = 0.0f;
  for (int c = 0; c < Ch; ++c) mean += src[c];
  mean /= (float)Ch;
  float var = 0.0f;
  for (int c = 0; c < Ch; ++c) { float d = src[c] - mean; var += d * d; }
  var /= (float)Ch;
  float r = rsqrtf(var + 1e-5f);
  float* yo = Y + (size_t)idx * (2 * Ch);
  for (int c = 0; c < Ch; ++c) yo[c] = (src[c] - mean) * r * g[c] + bt[c];
  const float* sk = SK + (size_t)idx * Ch;
  for (int c = 0; c < Ch; ++c) yo[Ch + c] = sk[c];
}

// ---------------------------------------------------------------------------
// Final 2x expansion rearrange: YT(B,56,56,192) -> (B,112,112,48) + bias + LN.
// Feature o of output (2h+k,2w+l) sits at YT[...,(o*4 + k*2 + l)].
// ---------------------------------------------------------------------------
__global__ void final_ln_kernel(const float* __restrict__ YT,
                                const float* __restrict__ fb,
                                const float* __restrict__ g,
                                const float* __restrict__ bt,
                                float* __restrict__ Y, int Bz) {
  const int H = 56, W = 56, HO = 112, WO = 112, CO = 48;
  int total = Bz * HO * WO;
  int idx = blockIdx.x * blockDim.x + threadIdx.x;
  if (idx >= total) return;
  int x = idx % WO;
  int t = idx / WO;
  int y = t % HO;
  int b = t / HO;
  int h = y >> 1, k = y & 1, w = x >> 1, l = x & 1;
  const float* src = YT + (((size_t)b * H + h) * W + w) * 192 + (k * 2 + l);
  float v[CO];
  float mean = 0.0f;
#pragma unroll
  for (int o = 0; o < CO; ++o) { v[o] = src[o * 4] + fb[o]; mean += v[o]; }
  mean /= (float)CO;
  float var = 0.0f;
#pragma unroll
  for (int o = 0; o < CO; ++o) { float d = v[o] - mean; var += d * d; }
  var /= (float)CO;
  float r = rsqrtf(var + 1e-5f);
  float* y_out = Y + (size_t)idx * CO;
#pragma unroll
  for (int o = 0; o < CO; ++o) y_out[o] = (v[o] - mean) * r * g[o] + bt[o];
}

// ---------------------------------------------------------------------------
// Bilinear resize (B,112,112,21) channel-last -> (B,21,480,480), half-pixel.
// ---------------------------------------------------------------------------
__global__ void resize_kernel(const float* __restrict__ Yin,
                              float* __restrict__ Out, int Bz) {
  const int HI = 112, HO = 480;
  int total = Bz * NCLS * HO * HO;
  int idx = blockIdx.x * blockDim.x + threadIdx.x;
  if (idx >= total) return;
  int ox = idx % HO;
  int t = idx / HO;
  int oy = t % HO; t /= HO;
  int c = t % NCLS;
  int b = t / NCLS;
  const float s = (float)HI / (float)HO;
  float fy = (oy + 0.5f) * s - 0.5f;
  float fx = (ox + 0.5f) * s - 0.5f;
  float yf = floorf(fy), xf = floorf(fx);
  float wy = fy - yf, wx = fx - xf;
  int y0 = (int)yf, x0 = (int)xf;
  int y1 = y0 + 1, x1 = x0 + 1;
  y0 = min(max(y0, 0), HI - 1); y1 = min(max(y1, 0), HI - 1);
  x0 = min(max(x0, 0), HI - 1); x1 = min(max(x1, 0), HI - 1);
  const float* base = Yin + (size_t)b * HI * HI * NCLS;
  float v00 = base[((size_t)y0 * HI + x0) * NCLS + c];
  float v01 = base[((size_t)y0 * HI + x1) * NCLS + c];
  float v10 = base[((size_t)y1 * HI + x0) * NCLS + c];
  float v11 = base[((size_t)y1 * HI + x1) * NCLS + c];
  Out[idx] = (1.0f - wy) * ((1.0f - wx) * v00 + wx * v01) +
             wy * ((1.0f - wx) * v10 + wx * v11);
}

// ---------------------------------------------------------------------------
// Host-side orchestration
// ---------------------------------------------------------------------------
static inline int cdiv(int a, int b) { return (a + b - 1) / b; }

static float* g_wp = nullptr;   // per-call repack buffer (set in kernel_launch)

static void launch_gemm(const float* A, const float* W, const float* bias,
                        float* C, int M, int N, int K, hipStream_t s) {
  int rp = K * N;
  repack_w_kernel<<<cdiv(rp, 256), 256, 0, s>>>(W, g_wp, N, K);
  int tiles = cdiv(M, 16) * cdiv(N, 32);
  int blocks = cdiv(tiles, 8);  // 8 waves (256 threads) per block
  gemm_wmma_f32_kernel<<<blocks, 256, 0, s>>>(A, g_wp, bias, C, M, N, K);
}

struct NaP {
  const float *qw, *qb, *kw, *kb, *vw, *vb, *ow, *ob, *rb;
};
static NaP get_na(void* const* d_in, int base) {
  NaP p;
  p.qw = (const float*)d_in[base + 0]; p.qb = (const float*)d_in[base + 1];
  p.kw = (const float*)d_in[base + 2]; p.kb = (const float*)d_in[base + 3];
  p.vw = (const float*)d_in[base + 4]; p.vb = (const float*)d_in[base + 5];
  p.ow = (const float*)d_in[base + 6]; p.ob = (const float*)d_in[base + 7];
  p.rb = (const float*)d_in[base + 8];
  return p;
}

static void run_na(const float* src, float* dst, const NaP& p, int Bz, int N,
                   int C, float* Qb, float* Kb, float* Vb, float* Sb,
                   hipStream_t s) {
  int M = Bz * N;
  launch_gemm(src, p.qw, p.qb, Qb, M, C, C, s);
  launch_gemm(src, p.kw, p.kb, Kb, M, C, C, s);
  launch_gemm(src, p.vw, p.vb, Vb, M, C, C, s);
  float scale = 1.0f / sqrtf((float)C);
  int tot = M * KNB;
  na_scores_kernel<<<cdiv(tot, 256), 256, 0, s>>>(Qb, Kb, p.rb, Sb, Bz, N, C, scale);
  na_softmax_kernel<<<cdiv(M, 256), 256, 0, s>>>(Sb, M);
  int tot2 = M * C;
  na_out_kernel<<<cdiv(tot2, 256), 256, 0, s>>>(Sb, Vb, Qb, Bz, N, C);  // reuse Qb
  launch_gemm(Qb, p.ow, p.ob, dst, M, C, C, s);
}

extern "C" void kernel_launch(void* const* d_in, const int* in_sizes, int n_in,
                              void* d_out, int out_size, void* d_ws,
                              size_t ws_size, hipStream_t stream) {
  const int B = 8;
  // ----- input map (setup_inputs dict insertion order) -----
  // 0:x 1:embed.w 2:embed.b 3..11:enc0 12..20:enc1 21..29:enc2
  // 30..32:merge0(ln.g,ln.b,red) 33..35:merge1 36..38:merge2
  // 39..47:bot 48..50:expand0(w,ln.g,ln.b) 51..53:expand1 54..56:expand2
  // 57..62:skip(w,b x3) 63..71:dec0 72..80:dec1 81..89:dec2
  // 90:final.w 91:final.b 92:final.ln.g 93:final.ln.b 94:head.w 95:head.b
  const float* x     = (const float*)d_in[0];
  const float* emb_w = (const float*)d_in[1];
  const float* emb_b = (const float*)d_in[2];
  NaP enc[3] = {get_na(d_in, 3), get_na(d_in, 12), get_na(d_in, 21)};
  const float* mg_g[3] = {(const float*)d_in[30], (const float*)d_in[33], (const float*)d_in[36]};
  const float* mg_b[3] = {(const float*)d_in[31], (const float*)d_in[34], (const float*)d_in[37]};
  const float* mg_r[3] = {(const float*)d_in[32], (const float*)d_in[35], (const float*)d_in[38]};
  NaP bot = get_na(d_in, 39);
  const float* ex_w[3] = {(const float*)d_in[48], (const float*)d_in[51], (const float*)d_in[54]};
  const float* ex_g[3] = {(const float*)d_in[49], (const float*)d_in[52], (const float*)d_in[55]};
  const float* ex_b[3] = {(const float*)d_in[50], (const float*)d_in[53], (const float*)d_in[56]};
  const float* sk_w[3] = {(const float*)d_in[57], (const float*)d_in[59], (const float*)d_in[61]};
  const float* sk_b[3] = {(const float*)d_in[58], (const float*)d_in[60], (const float*)d_in[62]};
  NaP dec[3] = {get_na(d_in, 63), get_na(d_in, 72), get_na(d_in, 81)};
  const float* fin_w  = (const float*)d_in[90];
  const float* fin_b  = (const float*)d_in[91];
  const float* fin_g  = (const float*)d_in[92];
  const float* fin_bt = (const float*)d_in[93];
  const float* head_w = (const float*)d_in[94];
  const float* head_b = (const float*)d_in[95];

  // ----- workspace carve (floats) -----
  float* ws = (float*)d_ws;
  size_t off = 0;
  auto carve = [&](size_t n) { float* p = ws + off; off += n; return p; };
  float* A0 = carve(6u * 1024 * 1024);
  float* A1 = carve(6u * 1024 * 1024);
  float* Qb = carve(3u * 1024 * 1024);
  float* Kb = carve(3u * 1024 * 1024);
  float* Vb = carve(3u * 1024 * 1024);
  float* Sb = carve(512u * 1024);
  float* sk0 = carve(2500000);     // (8,56,56,96)
  float* sk1 = carve(1300000);     // (8,28,28,192)
  float* sk2 = carve(700000);      // (8,14,14,384)
  g_wp = carve(1310720);           // weight repack: max K*N = 1536*768
  float* skbuf[3] = {sk0, sk1, sk2};
  (void)ws_size; (void)in_sizes; (void)n_in; (void)out_size;

  const int Hs[4] = {56, 28, 14, 7};
  const int Cs[4] = {96, 192, 384, 768};

  // ----- patch embed -----
  {
    int tot = B * 56 * 56 * 96;
    patch_embed_kernel<<<cdiv(tot, 256), 256, 0, stream>>>(x, emb_w, emb_b, A0, B);
  }
  float* cur = A0;

  // ----- encoder -----
  for (int i = 0; i < 3; ++i) {
    int H = Hs[i], C = Cs[i], N = H * H;
    run_na(cur, skbuf[i], enc[i], B, N, C, Qb, Kb, Vb, Sb, stream);
    int tot = B * (H / 2) * (H / 2);
    merge_ln_kernel<<<cdiv(tot, 256), 256, 0, stream>>>(skbuf[i], mg_g[i], mg_b[i],
                                                        A0, B, H, H, C);
    launch_gemm(A0, mg_r[i], nullptr, A1, tot, 2 * C, 4 * C, stream);
    cur = A1;
  }

  // ----- bottleneck (8,7,7,768) -----
  run_na(cur, A0, bot, B, 49, 768, Qb, Kb, Vb, Sb, stream);
  cur = A0;

  // ----- decoder -----
  for (int i = 0; i < 3; ++i) {
    int H = Hs[3 - i];       // 7, 14, 28
    int C = Cs[3 - i];       // 768, 384, 192
    int M = B * H * H;
    launch_gemm(cur, ex_w[i], nullptr, A1, M, 2 * C, C, stream);
    int tot = B * (2 * H) * (2 * H);
    expand_ln_concat_kernel<<<cdiv(tot, 256), 256, 0, stream>>>(
        A1, ex_g[i], ex_b[i], skbuf[2 - i], A0, B, H, H, C);
    launch_gemm(A0, sk_w[i], sk_b[i], A1, tot, C / 2, C, stream);
    run_na(A1, A0, dec[i], B, (2 * H) * (2 * H), C / 2, Qb, Kb, Vb, Sb, stream);
    cur = A0;
  }

  // ----- final: (8,56,56,96) -> GEMM(96->192) -> rearrange+LN -> head -> resize
  {
    int M = B * 56 * 56;  // 25088
    launch_gemm(cur, fin_w, nullptr, A1, M, 192, 96, stream);  // fin_w is (96,192) row-major
    int tot = B * 112 * 112;
    final_ln_kernel<<<cdiv(tot, 256), 256, 0, stream>>>(A1, fin_b, fin_g, fin_bt, A0, B);
    launch_gemm(A0, head_w, head_b, A1, tot, NCLS, 48, stream);  // (B,112,112,21)
    int rt = B * NCLS * 480 * 480;
    resize_kernel<<<cdiv(rt, 256), 256, 0, stream>>>(A1, (float*)d_out, B);
  }
}